// DiffNet_17085379903955
// MI455X (gfx1250) — compile-verified
//
#include <hip/hip_runtime.h>

typedef __attribute__((ext_vector_type(2))) float v2f;
typedef __attribute__((ext_vector_type(8))) float v8f;

#define RATE   0.01f
#define BROWS  64
#define KC     64
#define STRIDE 68   // 64 + 4 pad: 16B-aligned rows, bank-conflict-free ds_load_b64 frags

#define A_ELEMS (BROWS * STRIDE)   // one A buffer, floats
#define W_ELEMS (32 * STRIDE)      // one W buffer, floats

// One fused layer: out[b,n] = vj + scale*(c0*s2[b] + c1*z + c2*vj*s1[b] + bm*s1[b])
// with z = vi@W.T, vj = relu(z + fc_b).  GEMM via V_WMMA_F32_16X16X4_F32 with
// double-buffered async global->LDS staging (ASYNCcnt pipelining).
__global__ __launch_bounds__(256) void diffnet_layer(
    const float* __restrict__ in,   // [64, K]  (vi)
    const float* __restrict__ W,    // [N, K]   row-major
    const float* __restrict__ fcb,  // [N]
    float* __restrict__ out,        // [64, N]
    int K, int N,
    const float* __restrict__ c1w, const float* __restrict__ c1b,
    const float* __restrict__ c2w, const float* __restrict__ c2b,
    const int* __restrict__ bnum)
{
    __shared__ float Alds[2 * A_ELEMS];   // double-buffered 64x64 fp32 activation chunk
    __shared__ float Wlds[2 * W_ELEMS];   // double-buffered 32x64 fp32 weight chunk
    __shared__ float s1s[BROWS];
    __shared__ float s2s[BROWS];

    const int tid  = threadIdx.x;
    const int lane = tid & 31;
    const int wid  = tid >> 5;   // wave 0..7
    const int hi   = lane >> 4;  // half-wave: selects K pair {0,1} vs {2,3}
    const int lm   = lane & 15;  // row/col inside 16x16 tile

    // Wave-relative LDS byte offsets (flat-pointer low 32 bits = LDS offset, ISA §10.2).
    const unsigned AldsBase = (unsigned)(size_t)&Alds[0];
    const unsigned WldsBase = (unsigned)(size_t)&Wlds[0];

    // Collapse the meta-MLP: coef = conv2_w @ conv1_w (3 vals), bm = conv2_w@conv1_b + conv2_b
    float c0 = 0.f, c1 = 0.f, c2 = 0.f, bm = 0.f;
    #pragma unroll
    for (int h = 0; h < 8; ++h) {
        float w2 = c2w[h];
        c0 += w2 * c1w[h * 3 + 0];
        c1 += w2 * c1w[h * 3 + 1];
        c2 += w2 * c1w[h * 3 + 2];
        bm += w2 * c1b[h];
    }
    bm += c2b[0];
    const float scale = RATE / (float)bnum[0];

    // Issue one chunk's async copies: 4 b128 for A (64x64) + 2 b128 for W (32x64) per thread.
    auto issue_chunk = [&](int kc, int buf, int nbase) {
        #pragma unroll
        for (int i = 0; i < 4; ++i) {
            int idx = tid + i * 256;
            int row = idx >> 4;
            int kin = (idx & 15) << 2;
            unsigned ldsa = AldsBase + (unsigned)((buf * A_ELEMS + row * STRIDE + kin) * 4);
            unsigned goff = (unsigned)((row * K + kc + kin) * 4);
            asm volatile("global_load_async_to_lds_b128 %0, %1, %2"
                         :: "v"(ldsa), "v"(goff), "s"(in) : "memory");
        }
        #pragma unroll
        for (int i = 0; i < 2; ++i) {
            int idx = tid + i * 256;
            int row = idx >> 4;
            int kin = (idx & 15) << 2;
            unsigned ldsw = WldsBase + (unsigned)((buf * W_ELEMS + row * STRIDE + kin) * 4);
            unsigned goff = (unsigned)(((nbase + row) * K + kc + kin) * 4);
            asm volatile("global_load_async_to_lds_b128 %0, %1, %2"
                         :: "v"(ldsw), "v"(goff), "s"(W) : "memory");
        }
    };

    // Per-row s1 = sum(vi), s2 = sum(vi^2): each wave reduces 8 rows (wave32 butterflies).
    for (int r = wid * 8; r < wid * 8 + 8; ++r) {
        float a = 0.f, b = 0.f;
        for (int k = lane; k < K; k += 32) {
            float v = in[r * K + k];
            a += v;
            b += v * v;
        }
        #pragma unroll
        for (int off = 16; off > 0; off >>= 1) {
            a += __shfl_xor(a, off);
            b += __shfl_xor(b, off);
        }
        if (lane == 0) { s1s[r] = a; s2s[r] = b; }
    }

    // Tile assignment: block owns 64 rows x 32 cols; wave -> one 16x16 tile.
    const int nbase = blockIdx.x * 32;
    const int m0  = (wid & 3) * 16;
    const int n0l = (wid >> 2) * 16;

    v8f acc = {0.f, 0.f, 0.f, 0.f, 0.f, 0.f, 0.f, 0.f};

    const int nch = K / KC;
    issue_chunk(0, 0, nbase);   // prologue: chunk 0 -> buffer 0

    for (int c = 0; c < nch; ++c) {
        const int cur = c & 1;
        const int nxt = cur ^ 1;

        // (1) All waves finished reading buf[nxt] (consumed at iteration c-1):
        //     safe to start overwriting it with chunk c+1.
        __syncthreads();
        if (c + 1 < nch) {
            issue_chunk((c + 1) * KC, nxt, nbase);
            // 6 new ops outstanding; asyncs retire in order -> chunk c's 6 are done.
            asm volatile("s_wait_asynccnt 0x6" ::: "memory");
        } else {
            asm volatile("s_wait_asynccnt 0x0" ::: "memory");
        }
        // (2) Every wave waited on its own copies for chunk c -> LDS data visible to all.
        __syncthreads();

        // Fragment pointers per ISA fp32 A/B layout:
        // lane holds M(or N)=lm, K = kk + hi*2 + {0,1} in the two fragment dwords.
        const float* ap = &Alds[cur * A_ELEMS + (m0  + lm) * STRIDE + hi * 2];
        const float* bp = &Wlds[cur * W_ELEMS + (n0l + lm) * STRIDE + hi * 2];
        #pragma unroll
        for (int kk = 0; kk < KC; kk += 4) {
            v2f a, b;
            a[0] = ap[kk]; a[1] = ap[kk + 1];
            b[0] = bp[kk]; b[1] = bp[kk + 1];
            acc = __builtin_amdgcn_wmma_f32_16x16x4_f32(
                false, a, false, b, (short)0, acc, false, false);
        }
    }

    // Epilogue: D-matrix layout -> lane lm = column, vgpr v + hi*8 = row.
    const int col = nbase + n0l + lm;
    const float biasn = fcb[col];
    #pragma unroll
    for (int v = 0; v < 8; ++v) {
        int row  = m0 + v + hi * 8;
        float z  = acc[v];
        float vj = fmaxf(z + biasn, 0.f);
        float r  = vj + scale * (c0 * s2s[row] + c1 * z + (c2 * vj + bm) * s1s[row]);
        out[row * N + col] = r;
    }
}

extern "C" void kernel_launch(void* const* d_in, const int* in_sizes, int n_in,
                              void* d_out, int out_size, void* d_ws, size_t ws_size,
                              hipStream_t stream) {
    const float* x    = (const float*)d_in[0];
    const float* fc1w = (const float*)d_in[1];
    const float* fc1b = (const float*)d_in[2];
    const float* fc2w = (const float*)d_in[3];
    const float* fc2b = (const float*)d_in[4];
    const float* fc3w = (const float*)d_in[5];
    const float* fc3b = (const float*)d_in[6];
    const float* c1w  = (const float*)d_in[7];
    const float* c1b  = (const float*)d_in[8];
    const float* c2w  = (const float*)d_in[9];
    const float* c2b  = (const float*)d_in[10];
    const int*   bnum = (const int*)d_in[11];

    float* act1 = (float*)d_ws;        // [64,512] = 128 KB
    float* act2 = act1 + 64 * 512;     // [64,512] = 128 KB
    float* outp = (float*)d_out;       // [64,256]

    diffnet_layer<<<dim3(512 / 32), dim3(256), 0, stream>>>(
        x,    fc1w, fc1b, act1, 1024, 512, c1w, c1b, c2w, c2b, bnum);
    diffnet_layer<<<dim3(512 / 32), dim3(256), 0, stream>>>(
        act1, fc2w, fc2b, act2,  512, 512, c1w, c1b, c2w, c2b, bnum);
    diffnet_layer<<<dim3(256 / 32), dim3(256), 0, stream>>>(
        act2, fc3w, fc3b, outp,  512, 256, c1w, c1b, c2w, c2b, bnum);
}